// Model_11879879543162
// MI455X (gfx1250) — compile-verified
//
#include <hip/hip_runtime.h>
#include <hip/hip_bf16.h>
#include <math.h>

// Problem constants (match reference)
#define BB   128
#define TT   257        // T = U+1
#define VV   1024
#define UU   256
#define DD   512        // D_EMB
#define SS   513        // 2U+1
#define NEGV (-1.0e30f)
#define MSTRIDE 258     // LDS row stride for M (odd-ish, conflict-friendly)

typedef __attribute__((ext_vector_type(2))) float v2f;
typedef __attribute__((ext_vector_type(8))) float v8f;

__global__ void zero_out_kernel(float* out) { out[0] = 0.0f; }

// One workgroup per batch element. 512 threads = 16 waves (wave32).
// Dynamic LDS: M[TT][MSTRIDE] floats (the 257x257 selected-logit matrix).
__global__ __launch_bounds__(512) void ctc_fused_kernel(
    const float* __restrict__ x,       // (B, T, V)
    const int*   __restrict__ xlen,    // (B,)
    const int*   __restrict__ y,       // (B, U)
    const int*   __restrict__ ylen,    // (B,)
    const float* __restrict__ embed,   // (N_EMB, D)
    const float* __restrict__ linw,    // (V, D)
    const float* __restrict__ linb,    // (V,)
    float* __restrict__ out)
{
    extern __shared__ float Mlds[];            // TT * MSTRIDE floats
    __shared__ int           ids[TT];          // [0, y_eff[0..U-1]]  (A rows AND B cols)
    __shared__ float         lbg[TT];          // gathered bias
    __shared__ unsigned char skipok[SS + 3];
    __shared__ float         alpha[2][SS];

    const int b    = blockIdx.x;
    const int tid  = threadIdx.x;
    const int wave = tid >> 5;
    const int lane = tid & 31;
    const int hh   = lane >> 4;                // lane half (0/1)
    const int nn   = lane & 15;

    const int xl = xlen[b];
    const int yl = ylen[b];

    // ---- Phase 0a: build id list (blank + pad-masked labels) + bias gather ----
    for (int i = tid; i < TT; i += 512) {
        int v = 0;
        if (i > 0) {
            int lab = y[b * UU + (i - 1)];
            if ((i - 1) >= yl) lab = 0;        // pad mask
            v = lab;
        }
        ids[i] = v;
        lbg[i] = linb[v];
    }
    __syncthreads();

    // ---- Phase 0b: CTC skip flags (uses ids) ----
    for (int s = tid; s < SS; s += 512) {
        unsigned char ok = 0;
        if (s & 1) {
            int e   = ids[(s + 1) >> 1];
            int em2 = (s >= 3) ? ids[(s - 1) >> 1] : -1;
            ok = (e != 0 && e != em2) ? 1 : 0;
        }
        skipok[s] = ok;
    }

    // ---- Phase 1: WMMA GEMM  M = H_b (257x512) * W_sel^T (512x257) + x + bias ----
    // 17x17 tiles of 16x16 distributed across 16 waves.
    for (int tile = wave; tile < 17 * 17; tile += 16) {
        const int ti = tile / 17;
        const int tj = tile % 17;

        int ar = 16 * ti + nn;  if (ar > TT - 1) ar = TT - 1;   // clamped A row (t)
        int bc = 16 * tj + nn;  if (bc > TT - 1) bc = TT - 1;   // clamped B col (u)

        const float* __restrict__ Ap = embed + (size_t)ids[ar] * DD + 2 * hh;
        const float* __restrict__ Bp = linw  + (size_t)ids[bc] * DD + 2 * hh;

        v8f c = {};
#pragma unroll 8
        for (int k = 0; k < DD; k += 4) {
            // A 16x4 f32 layout: lane holds K = {k,k+1}+2h  (contiguous float2)
            v2f af = *(const v2f*)(Ap + k);
            // B 4x16 f32 layout mirrored: lane holds K = {k,k+1}+2h at col n
            v2f bf = *(const v2f*)(Bp + k);
            c = __builtin_amdgcn_wmma_f32_16x16x4_f32(
                    false, af, false, bf, (short)0, c, false, false);
        }

        // Epilogue: add x-gather + bias, store valid region of tile into LDS M.
        const int u = 16 * tj + nn;
        if (u < TT) {
            const int   idu = ids[u];
            const float lb  = lbg[u];
            const float* xb = x + ((size_t)b * TT) * VV + idu;
#pragma unroll
            for (int e = 0; e < 8; ++e) {
                const int t = 16 * ti + e + 8 * hh;
                if (t < TT) {
                    Mlds[t * MSTRIDE + u] = c[e] + xb[(size_t)t * VV] + lb;
                }
            }
        }
    }
    __syncthreads();

    // ---- Phase 2: CTC forward recursion, entirely in LDS ----
    // lp_ext[t,s] = M[t][col(s)],  col(s) = s odd ? (s+1)/2 : 0 (blank)
    for (int s = tid; s < SS; s += 512) {
        float v = NEGV;
        if (s == 0) v = Mlds[0];
        else if (s == 1) v = Mlds[1];
        alpha[0][s] = v;
    }
    __syncthreads();

    int cur = 0;
    for (int t = 1; t < xl; ++t) {
        const float* __restrict__ Mt = Mlds + (size_t)t * MSTRIDE;
        for (int s = tid; s < SS; s += 512) {
            float a0 = alpha[cur][s];
            float a1 = (s >= 1) ? alpha[cur][s - 1] : NEGV;
            float a2 = (s >= 2 && skipok[s]) ? alpha[cur][s - 2] : NEGV;
            float m  = fmaxf(a0, fmaxf(a1, a2));
            float l  = m + logf(expf(a0 - m) + expf(a1 - m) + expf(a2 - m));
            int col  = (s & 1) ? ((s + 1) >> 1) : 0;
            alpha[cur ^ 1][s] = Mt[col] + l;
        }
        __syncthreads();
        cur ^= 1;
    }

    if (tid == 0) {
        const int sl = 2 * yl;
        float a = alpha[cur][sl];
        float bgm = alpha[cur][sl - 1];
        float m = fmaxf(a, bgm);
        float ll = m + logf(expf(a - m) + expf(bgm - m));
        atomicAdd(out, -ll);
    }
}

extern "C" void kernel_launch(void* const* d_in, const int* in_sizes, int n_in,
                              void* d_out, int out_size, void* d_ws, size_t ws_size,
                              hipStream_t stream) {
    (void)in_sizes; (void)n_in; (void)out_size; (void)d_ws; (void)ws_size;

    const float* x      = (const float*)d_in[0];
    const int*   xlen   = (const int*)  d_in[1];
    const int*   y      = (const int*)  d_in[2];
    const int*   ylen   = (const int*)  d_in[3];
    const float* embedw = (const float*)d_in[4];
    const float* linw   = (const float*)d_in[5];
    const float* linb   = (const float*)d_in[6];
    float*       out    = (float*)d_out;

    zero_out_kernel<<<1, 1, 0, stream>>>(out);

    const size_t smem = (size_t)TT * MSTRIDE * sizeof(float);   // ~265 KB of 320 KB LDS
    ctc_fused_kernel<<<BB, 512, smem, stream>>>(x, xlen, y, ylen, embedw, linw, linb, out);
}